// SPLLoss_76922864272129
// MI455X (gfx1250) — compile-verified
//
#include <hip/hip_runtime.h>
#include <math.h>

#define B_ROWS 16384          // power of two -> % becomes &
#define B_MASK (B_ROWS - 1)
#define C_COLS 4096
#define TPB    256            // 8 wave32 per block
#define NWAVE  (TPB / 32)

// Combine two (max, sum-of-exp) partials: S = s1*e^(m1-M) + s2*e^(m2-M)
__device__ __forceinline__ void combine_ms(float& m, float& s, float m2, float s2) {
    float M = fmaxf(m, m2);
    s = s * __expf(m - M) + s2 * __expf(m2 - M);
    m = M;
}

// Pass 1: deterministic last-write-wins via order-independent atomicMax on writer id.
__global__ void k_scatter_winner(const int* __restrict__ index,
                                 unsigned* __restrict__ winner) {
    int i = blockIdx.x * blockDim.x + threadIdx.x;
    unsigned pos = ((unsigned)index[i]) & B_MASK;     // B is power of two
    atomicMax(&winner[pos], (unsigned)(i + 1));
}

// Pass 2: only the winning writer materializes w[pos] = v[index[i]].
__global__ void k_build_w(const int* __restrict__ index, const float* __restrict__ v,
                          const unsigned* __restrict__ winner,
                          float* __restrict__ w) {
    int i = blockIdx.x * blockDim.x + threadIdx.x;
    unsigned pos = ((unsigned)index[i]) & B_MASK;
    if (winner[pos] == (unsigned)(i + 1)) w[pos] = v[index[i]];
}

// Main kernel: one block per row. Stage the 16KB row into LDS with the CDNA5
// async copy engine (global_load_async_to_lds_b128, ASYNCcnt-tracked,
// non-temporal since the 256MB stream has zero reuse), then single-pass
// online softmax. Rows with zero weight skip the 16KB read entirely.
__global__ void __launch_bounds__(TPB)
k_row_ce(const float* __restrict__ input, const int* __restrict__ target,
         const float* __restrict__ w, float* __restrict__ loss) {
    __shared__ __align__(16) float lds[C_COLS];
    __shared__ float wm[NWAVE], wsum[NWAVE];

    const int row = blockIdx.x;
    const int t   = threadIdx.x;

    const float wr = w[row];                 // uniform scalar load (L2-hot, 64KB)
    if (wr == 0.0f) {                        // whole-block early exit (uniform)
        if (t == 0) loss[row] = 0.0f;
        return;
    }

    const float* rowp = input + (size_t)row * C_COLS;

    // Async-copy 16KB row -> LDS: 4 rounds, each thread moves one float4.
    // Linear mapping: element e = r*1024 + t*4 (coalesced 4KB bursts per round,
    // conflict-free b128 LDS layout on readback).
    #pragma unroll
    for (int r = 0; r < 4; ++r) {
        const int e = r * (TPB * 4) + t * 4;
        unsigned laddr = (unsigned)(unsigned long long)(const void*)(lds + e);
        unsigned long long gaddr = (unsigned long long)(const void*)(rowp + e);
        asm volatile("global_load_async_to_lds_b128 %0, %1, off th:TH_LOAD_NT"
                     :: "v"(laddr), "v"(gaddr) : "memory");
    }
    asm volatile("s_wait_asynccnt 0x0" ::: "memory");  // own wave's copies done
    __syncthreads();                                   // all waves' copies done

    // Per-thread: max then sum-exp over 16 elements (4 x ds b128 reads).
    float x[16];
    #pragma unroll
    for (int r = 0; r < 4; ++r) {
        const float4 q = *(const float4*)(lds + r * (TPB * 4) + t * 4);
        x[r*4+0] = q.x; x[r*4+1] = q.y; x[r*4+2] = q.z; x[r*4+3] = q.w;
    }
    float m = x[0];
    #pragma unroll
    for (int k = 1; k < 16; ++k) m = fmaxf(m, x[k]);
    float s = 0.0f;
    #pragma unroll
    for (int k = 0; k < 16; ++k) s += __expf(x[k] - m);

    // wave32 butterfly merge of (m, s)
    #pragma unroll
    for (int off = 16; off >= 1; off >>= 1) {
        float m2 = __shfl_xor(m, off, 32);
        float s2 = __shfl_xor(s, off, 32);
        combine_ms(m, s, m2, s2);
    }
    const int wave = t >> 5;
    if ((t & 31) == 0) { wm[wave] = m; wsum[wave] = s; }
    __syncthreads();

    if (t == 0) {
        float M = wm[0], S = wsum[0];
        #pragma unroll
        for (int i = 1; i < NWAVE; ++i) combine_ms(M, S, wm[i], wsum[i]);
        const int tgt = target[row];
        const float xt = lds[tgt];           // free gather from staged row
        // CE = -(x_t - M - log(S)) ; weighted
        loss[row] = wr * (M + logf(S) - xt);
    }
}

// Deterministic final mean: fixed per-thread serial sums + fixed LDS tree.
__global__ void __launch_bounds__(TPB)
k_reduce(const float* __restrict__ loss, float* __restrict__ out) {
    __shared__ float sm[TPB];
    float s = 0.0f;
    for (int k = threadIdx.x; k < B_ROWS; k += TPB) s += loss[k];
    sm[threadIdx.x] = s;
    __syncthreads();
    for (int off = TPB / 2; off > 0; off >>= 1) {
        if (threadIdx.x < off) sm[threadIdx.x] += sm[threadIdx.x + off];
        __syncthreads();
    }
    if (threadIdx.x == 0) out[0] = sm[0] * (1.0f / (float)B_ROWS);
}

extern "C" void kernel_launch(void* const* d_in, const int* in_sizes, int n_in,
                              void* d_out, int out_size, void* d_ws, size_t ws_size,
                              hipStream_t stream) {
    const float* input  = (const float*)d_in[0];   // [B, C] f32
    const float* v      = (const float*)d_in[1];   // [N] f32
    const int*   target = (const int*)d_in[2];     // [B] int
    const int*   index  = (const int*)d_in[3];     // [B] int
    float* out = (float*)d_out;

    unsigned* winner = (unsigned*)d_ws;                                  // B u32
    float*    w      = (float*)((char*)d_ws + (size_t)B_ROWS * 4);       // B f32
    float*    loss   = (float*)((char*)d_ws + (size_t)B_ROWS * 8);       // B f32

    // zero winner + w each call (ws is not re-poisoned between replays)
    hipMemsetAsync(d_ws, 0, (size_t)B_ROWS * 8, stream);

    k_scatter_winner<<<B_ROWS / TPB, TPB, 0, stream>>>(index, winner);
    k_build_w      <<<B_ROWS / TPB, TPB, 0, stream>>>(index, v, winner, w);
    k_row_ce       <<<B_ROWS, TPB, 0, stream>>>(input, target, w, loss);
    k_reduce       <<<1, TPB, 0, stream>>>(loss, out);
}